// MinibatchDiscrimination_58179626991726
// MI455X (gfx1250) — compile-verified
//
#include <hip/hip_runtime.h>

typedef __attribute__((ext_vector_type(2))) float v2f;
typedef __attribute__((ext_vector_type(8))) float v8f;
typedef __attribute__((ext_vector_type(4))) unsigned int v4u;
typedef __attribute__((ext_vector_type(8))) int v8i;
typedef __attribute__((ext_vector_type(4))) int v4i;

constexpr int BATCH = 512;           // minibatch
constexpr int D     = 2048;          // input features
constexpr int OFEAT = 64;            // OUT_FEATURES
constexpr int KDIM  = 16;            // KERNEL_DIM
constexpr int N     = OFEAT * KDIM;  // 1024 GEMM columns
constexpr int OUTW  = D + OFEAT;     // 2112 output row width

// ---------------------------------------------------------------------------
// GEMM: m[BATCH, N] = x[BATCH, D] @ T[D, N], exact fp32 via V_WMMA_F32_16X16X4_F32
// One wave computes one 16x16 tile of m. 8 waves (256 threads) per block.
// ---------------------------------------------------------------------------
__global__ __launch_bounds__(256)
void gemm_wmma_f32(const float* __restrict__ x,
                   const float* __restrict__ T,
                   float* __restrict__ m) {
    const int lane = threadIdx.x & 31;
    const int wave = threadIdx.x >> 5;
    const int tile = blockIdx.x * 8 + wave;        // 0 .. 2047
    const int nTilesN = N / 16;                    // 64
    const int mt = tile / nTilesN;
    const int nt = tile % nTilesN;
    const int m0 = mt * 16;
    const int n0 = nt * 16;

    const int row  = lane & 15;                    // A row / B,C column selector
    const int kofs = (lane >> 4) << 1;             // 0 or 2 (k-pair within step)

    // A: lane loads x[m0+row][kk+kofs .. kk+kofs+1]  (contiguous float2)
    const float* __restrict__ arow = x + (size_t)(m0 + row) * D + kofs;
    // B: lane loads T[kk+kofs][n0+col], T[kk+kofs+1][n0+col]
    const float* __restrict__ bcol = T + (size_t)kofs * N + n0 + row;

    v8f acc = {};
    for (int kc = 0; kc < D; kc += 256) {
        // pull the next A chunk toward the WGP while this one is consumed
        __builtin_prefetch(arow + kc + 256, 0, 1);
#pragma unroll 8
        for (int kk = kc; kk < kc + 256; kk += 4) {
            v2f a;
            a.x = arow[kk];
            a.y = arow[kk + 1];
            v2f b;
            b.x = bcol[(size_t)kk * N];
            b.y = bcol[(size_t)(kk + 1) * N];
            // (neg_a, A, neg_b, B, c_mod, C, reuse_a, reuse_b)
            acc = __builtin_amdgcn_wmma_f32_16x16x4_f32(
                false, a, false, b, (short)0, acc, false, false);
        }
    }

    // C/D layout: VGPR r -> row m0 + r + (lane>>4)*8, col n0 + (lane&15)
    const int crow = m0 + ((lane >> 4) << 3);
    const int ccol = n0 + row;
#pragma unroll
    for (int r = 0; r < 8; ++r) {
        m[(size_t)(crow + r) * N + ccol] = acc[r];
    }
}

// ---------------------------------------------------------------------------
// Pairwise L1 distance over kernel dim + exp reduction.
// Block (o, chunk): stages m[:, o, :] (512x16 fp32 = 32 KB) into LDS with one
// Tensor Data Mover descriptor (tile 16 x 512, row stride 1024 elements),
// then each thread owns one i and sweeps all 512 j from LDS.
// out[i, D + o] = sum_j exp(-sum_k |m[i,o,k] - m[j,o,k]|)
// ---------------------------------------------------------------------------
__global__ __launch_bounds__(256)
void pairwise_l1_exp(const float* __restrict__ m,
                     float* __restrict__ out) {
    const int o = blockIdx.x;                      // 0 .. 63
    __shared__ float s[BATCH * KDIM];              // 32 KB of 320 KB WGP LDS

    if (threadIdx.x < 32) {                        // one wave issues the DMA
        // generic LDS pointer = {SHARED_BASE, 16'b0, offset} -> low 32 bits
        const unsigned int lds_off = (unsigned int)(uintptr_t)(void*)&s[0];
        const unsigned long long gaddr =
            (unsigned long long)(uintptr_t)(m + (size_t)o * KDIM);

        // --- D# group 0: count=1 | lds_addr | global_addr[56:0] | type=2 ---
        v4u g0;
        g0[0] = 1u;                                        // count=1, user D#
        g0[1] = lds_off;                                   // lds_addr (bytes)
        g0[2] = (unsigned int)gaddr;                       // global_addr[31:0]
        g0[3] = (unsigned int)((gaddr >> 32) & 0x01FFFFFFu)
              | 0x80000000u;                               // addr[56:32] | type=2
        // --- D# group 1: dims/strides in data_size units (4 B) ---
        v8i g1;
        g1[0] = 0x00020000;                // wg_mask=0, data_size=2 (4 bytes)
        g1[1] = (int)((unsigned)N << 16);  // tensor_dim0 = 1024 (low 16 -> [63:48])
        g1[2] = (int)(((unsigned)N >> 16) | ((unsigned)BATCH << 16)); // dim0 hi | dim1 lo
        g1[3] = (int)(((unsigned)BATCH >> 16) | ((unsigned)KDIM << 16)); // dim1 hi | tile_dim0=16
        g1[4] = BATCH;                     // tile_dim1 = 512, tile_dim2 = 0
        g1[5] = N;                         // tensor_dim0_stride = 1024 (low 32)
        g1[6] = 0;                         // stride hi | dim1_stride lo
        g1[7] = 0;
        v4i g2 = {0, 0, 0, 0};             // 2D tensor: groups 2/3 unused
        v4i g3 = {0, 0, 0, 0};
        v8i g4 = {0, 0, 0, 0, 0, 0, 0, 0}; // extra group (6-arg toolchain form)

        __builtin_amdgcn_tensor_load_to_lds(g0, g1, g2, g3, g4, 0);
        __builtin_amdgcn_s_wait_tensorcnt(0);
    }
    __syncthreads();

    const float NEG_LOG2E = -1.44269504088896340736f;
    const int i = (int)blockIdx.y * 256 + (int)threadIdx.x;   // one i per thread
    float mi[KDIM];
#pragma unroll
    for (int k = 0; k < KDIM; ++k) mi[k] = s[i * KDIM + k];

    float acc = 0.0f;
    for (int j = 0; j < BATCH; ++j) {               // s[j*16+k] is an LDS broadcast
        float d = 0.0f;
#pragma unroll
        for (int k = 0; k < KDIM; ++k) d += fabsf(mi[k] - s[j * KDIM + k]);
        acc += exp2f(d * NEG_LOG2E);                // exp(-d) on the TRANS pipe
    }
    out[(size_t)i * OUTW + D + o] = acc;
}

// ---------------------------------------------------------------------------
// Copy x into out[:, 0:2048] with float4 vectors (rows stay 16B aligned: 2112%4==0)
// ---------------------------------------------------------------------------
__global__ __launch_bounds__(256)
void copy_x(const float* __restrict__ x, float* __restrict__ out) {
    const int v = blockIdx.x * 256 + threadIdx.x;  // 0 .. 512*512-1 (float4 units)
    const int row  = v >> 9;                       // / (2048/4)
    const int col4 = v & 511;
    const float4 t = *(const float4*)(x + (size_t)row * D + (col4 << 2));
    *(float4*)(out + (size_t)row * OUTW + (col4 << 2)) = t;
}

extern "C" void kernel_launch(void* const* d_in, const int* in_sizes, int n_in,
                              void* d_out, int out_size, void* d_ws, size_t ws_size,
                              hipStream_t stream) {
    const float* x = (const float*)d_in[0];   // [512, 2048]
    const float* T = (const float*)d_in[1];   // [2048, 1024]
    float* out = (float*)d_out;               // [512, 2112]
    float* m   = (float*)d_ws;                // [512, 1024] scratch (2 MB)

    // 1) out[:, :2048] = x
    copy_x<<<(BATCH * D / 4) / 256, 256, 0, stream>>>(x, out);

    // 2) m = x @ T   (2048 16x16 tiles, 8 waves/block)
    gemm_wmma_f32<<<(BATCH / 16) * (N / 16) / 8, 256, 0, stream>>>(x, T, m);

    // 3) out[:, 2048+o] = sum_j exp(-L1(m[i,o,:], m[j,o,:]))
    pairwise_l1_exp<<<dim3(OFEAT, BATCH / 256), 256, 0, stream>>>(m, out);
}